// BaLu_IGMC_Imp_33827162423523
// MI455X (gfx1250) — compile-verified
//
#include <hip/hip_runtime.h>

// ---------------------------------------------------------------------------
// RGCN/GCN stack for MI455X (gfx1250).
// Dense node-feature x weight GEMMs use V_WMMA_F32_16X16X4_F32 (native f32
// WMMA, full precision). Edge aggregation uses f32 global atomics.
// ---------------------------------------------------------------------------

typedef __attribute__((ext_vector_type(2))) float v2f;
typedef __attribute__((ext_vector_type(8))) float v8f;

// One wave computes a 16x64 output tile of C[M x 64] = A[M x K] @ W[K x 64]
// (+bias)(+relu). K must be a multiple of 4 (here 128 or 64). M multiple of 16.
template <int BIAS, int RELU>
__global__ void gemm_n64_wmma(const float* __restrict__ A,
                              const float* __restrict__ W,
                              const float* __restrict__ bias,
                              float* __restrict__ C,
                              int M, int K) {
  const int lane = threadIdx.x & 31;
  const int wave = threadIdx.x >> 5;
  const int mt   = blockIdx.x * (blockDim.x >> 5) + wave;   // 16-row tile id
  const int row0 = mt << 4;
  if (row0 >= M) return;                                    // wave-uniform

  const int mrow  = row0 + (lane & 15);       // A row for this lane
  const int khalf = (lane >> 4) << 1;         // lanes 16-31 carry K+2,K+3
  const int ncol  = lane & 15;                // B/C column within 16-tile

  v8f acc0 = {}; v8f acc1 = {}; v8f acc2 = {}; v8f acc3 = {};
  const float* arow = A + (size_t)mrow * K;

  for (int k = 0; k < K; k += 4) {
    const int kb = k + khalf;                 // even -> 8B-aligned float2 load
    v2f a = *(const v2f*)(arow + kb);
    const float* w0 = W + (size_t)kb * 64 + ncol;
    v2f b0, b1, b2, b3;
    b0.x = w0[0];  b0.y = w0[64];             // n-tile 0  (cols  0..15)
    b1.x = w0[16]; b1.y = w0[80];             // n-tile 1  (cols 16..31)
    b2.x = w0[32]; b2.y = w0[96];             // n-tile 2  (cols 32..47)
    b3.x = w0[48]; b3.y = w0[112];            // n-tile 3  (cols 48..63)
    acc0 = __builtin_amdgcn_wmma_f32_16x16x4_f32(false, a, false, b0, (short)0, acc0, false, false);
    acc1 = __builtin_amdgcn_wmma_f32_16x16x4_f32(false, a, false, b1, (short)0, acc1, false, false);
    acc2 = __builtin_amdgcn_wmma_f32_16x16x4_f32(false, a, false, b2, (short)0, acc2, false, false);
    acc3 = __builtin_amdgcn_wmma_f32_16x16x4_f32(false, a, false, b3, (short)0, acc3, false, false);
  }

  float bb0 = 0.f, bb1 = 0.f, bb2 = 0.f, bb3 = 0.f;
  if (BIAS) {
    bb0 = bias[ncol];      bb1 = bias[16 + ncol];
    bb2 = bias[32 + ncol]; bb3 = bias[48 + ncol];
  }
  const int rbase = row0 + ((lane >> 4) << 3);   // lanes 16-31 hold rows +8
#pragma unroll
  for (int i = 0; i < 8; ++i) {
    float* crow = C + (size_t)(rbase + i) * 64 + ncol;
    float v0 = acc0[i] + bb0;
    float v1 = acc1[i] + bb1;
    float v2 = acc2[i] + bb2;
    float v3 = acc3[i] + bb3;
    if (RELU) {
      v0 = fmaxf(v0, 0.f); v1 = fmaxf(v1, 0.f);
      v2 = fmaxf(v2, 0.f); v3 = fmaxf(v3, 0.f);
    }
    crow[0] = v0; crow[16] = v1; crow[32] = v2; crow[48] = v3;
  }
}

// OUT[dst[e]][:] += T[src[e]][:]   (64 f32 channels, 16 threads per edge)
__global__ void scatter_add64(const float* __restrict__ T,
                              const int* __restrict__ src,
                              const int* __restrict__ dst,
                              float* __restrict__ OUT, int E) {
  int idx = blockIdx.x * blockDim.x + threadIdx.x;
  if (idx >= (E << 4)) return;
  int e = idx >> 4;
  int c = (idx & 15) << 2;
  int s = src[e], d = dst[e];
  const float4 v = *(const float4*)(T + (size_t)s * 64 + c);
  float* o = OUT + (size_t)d * 64 + c;
  unsafeAtomicAdd(o + 0, v.x);
  unsafeAtomicAdd(o + 1, v.y);
  unsafeAtomicAdd(o + 2, v.z);
  unsafeAtomicAdd(o + 3, v.w);
}

// Same, but only for edges whose relation type == r.
__global__ void scatter_add64_rel(const float* __restrict__ T,
                                  const int* __restrict__ src,
                                  const int* __restrict__ dst,
                                  const int* __restrict__ rel, int r,
                                  float* __restrict__ OUT, int E) {
  int idx = blockIdx.x * blockDim.x + threadIdx.x;
  if (idx >= (E << 4)) return;
  int e = idx >> 4;
  if (rel[e] != r) return;
  int c = (idx & 15) << 2;
  int s = src[e], d = dst[e];
  const float4 v = *(const float4*)(T + (size_t)s * 64 + c);
  float* o = OUT + (size_t)d * 64 + c;
  unsafeAtomicAdd(o + 0, v.x);
  unsafeAtomicAdd(o + 1, v.y);
  unsafeAtomicAdd(o + 2, v.z);
  unsafeAtomicAdd(o + 3, v.w);
}

__global__ void relu_inplace(float* __restrict__ p, int n4) {
  int i = blockIdx.x * blockDim.x + threadIdx.x;
  if (i >= n4) return;
  float4 v = ((float4*)p)[i];
  v.x = fmaxf(v.x, 0.f); v.y = fmaxf(v.y, 0.f);
  v.z = fmaxf(v.z, 0.f); v.w = fmaxf(v.w, 0.f);
  ((float4*)p)[i] = v;
}

extern "C" void kernel_launch(void* const* d_in, const int* in_sizes, int n_in,
                              void* d_out, int out_size, void* d_ws, size_t ws_size,
                              hipStream_t stream) {
  const float* x     = (const float*)d_in[0];
  const int*   eidx  = (const int*)d_in[1];
  const int*   reidx = (const int*)d_in[2];
  const int*   rtype = (const int*)d_in[3];
  const int N = in_sizes[0] / 128;   // 100000
  const int E = in_sizes[1] / 2;     // 1600000
  const int* src  = eidx;          const int* dst  = eidx + E;
  const int* rsrc = reidx;         const int* rdst = reidx + E;

  // Workspace: 3 buffers of N x 64 f32 (~77 MB total)
  float* bufA = (float*)d_ws;
  float* bufB = bufA + (size_t)N * 64;
  float* T    = bufB + (size_t)N * 64;
  float* buf[2] = {bufA, bufB};
  int bi = 0;

  const int mtiles = (N + 15) >> 4;
  dim3 gblock(128);                               // 4 waves, 1 M-tile each
  dim3 ggrid((mtiles + 3) / 4);
  dim3 sblock(256), sgrid(((E << 4) + 255) / 256);
  const int n4 = N * 16;                          // N*64/4 float4s
  dim3 rblock(256), rgrid((n4 + 255) / 256);

  auto gemm = [&](const float* A, const float* W, const float* b, float* C,
                  int K, int bias, int relu) {
    if (bias && relu)
      gemm_n64_wmma<1, 1><<<ggrid, gblock, 0, stream>>>(A, W, b, C, N, K);
    else if (bias)
      gemm_n64_wmma<1, 0><<<ggrid, gblock, 0, stream>>>(A, W, b, C, N, K);
    else
      gemm_n64_wmma<0, 0><<<ggrid, gblock, 0, stream>>>(A, W, nullptr, C, N, K);
  };

  const float* h = x;
  int K = 128;
  for (int l = 0; l < 3; ++l) {
    const int base = 4 + 6 * l;
    const float* gWs = (const float*)d_in[base + 0];
    const float* gWn = (const float*)d_in[base + 1];
    const float* gb  = (const float*)d_in[base + 2];
    const float* rW  = (const float*)d_in[base + 3];   // [5,64,64]
    const float* rWr = (const float*)d_in[base + 4];
    const float* rb  = (const float*)d_in[base + 5];

    // ---- GCN: o = relu(h@Ws + segsum((h@Wn)[src] -> dst) + b) ----
    float* o = buf[bi]; bi ^= 1;
    gemm(h, gWs, gb, o, K, /*bias=*/1, /*relu=*/0);   // self term (+b)
    gemm(h, gWn, nullptr, T, K, 0, 0);                // neighbor transform
    scatter_add64<<<sgrid, sblock, 0, stream>>>(T, src, dst, o, E);
    relu_inplace<<<rgrid, rblock, 0, stream>>>(o, n4);
    h = o; K = 64;

    // ---- RGCN: o = relu(h@Wroot + sum_r segsum((h@W_r)[rsrc] -> rdst) + b) --
    o = buf[bi]; bi ^= 1;
    gemm(h, rWr, rb, o, 64, 1, 0);                    // root term (+b)
    for (int r = 0; r < 5; ++r) {
      gemm(h, rW + (size_t)r * 64 * 64, nullptr, T, 64, 0, 0);
      scatter_add64_rel<<<sgrid, sblock, 0, stream>>>(T, rsrc, rdst, rtype, r, o, E);
    }
    relu_inplace<<<rgrid, rblock, 0, stream>>>(o, n4);
    h = o;
  }

  // ---- Output projection: d_out = relu(h @ Wout + bout) ----
  const float* Wout = (const float*)d_in[22];
  const float* bout = (const float*)d_in[23];
  gemm(h, Wout, bout, (float*)d_out, 64, 1, 1);
}